// CellLogitLSE_64819646432061
// MI455X (gfx1250) — compile-verified
//
#include <hip/hip_runtime.h>
#include <hip/hip_bf16.h>
#include <stdint.h>

#define NUM_CLASSES 32
#define MAX_ROWS    128            // counts are in [0,128)
#define WPB         2              // wave32 waves per block
#define NBUF        2              // double-buffered LDS staging per wave
#define BLOCKS      2048           // persistent grid: 4096 waves, ~4 images each
#define SCAN_THREADS 1024

typedef unsigned int u32x4 __attribute__((ext_vector_type(4)));
typedef int          i32x4 __attribute__((ext_vector_type(4)));
typedef int          i32x8 __attribute__((ext_vector_type(8)));
typedef __attribute__((address_space(3))) float lds_float;

#if defined(__AMDGCN__) && __has_builtin(__builtin_amdgcn_tensor_load_to_lds) && \
    __has_builtin(__builtin_amdgcn_s_wait_tensorcnt) && __has_builtin(__builtin_amdgcn_readfirstlane)
#define USE_TDM 1
#else
#define USE_TDM 0
#endif

#if defined(__AMDGCN__) && __has_builtin(__builtin_amdgcn_exp2f) && __has_builtin(__builtin_amdgcn_logf)
#define FAST_EXP2(x) __builtin_amdgcn_exp2f(x)
#define FAST_LOG2(x) __builtin_amdgcn_logf(x)
#else
#define FAST_EXP2(x) exp2f(x)
#define FAST_LOG2(x) log2f(x)
#endif

// ---------------------------------------------------------------------------
// Stage one segment (n rows x 32 f32, row-contiguous) into LDS.
// TDM path: one tensor_load_to_lds descriptor, tracked by TENSORcnt.
// Fallback: synchronous per-lane b128 copies (4 rows / 512B per iteration).
// ---------------------------------------------------------------------------
__device__ __forceinline__ void stage_tile(const float* __restrict__ src, int n,
                                           float* lbuf, int lane) {
#if USE_TDM
    // Wave-uniform descriptor material -> force into SGPRs.
    const uint32_t lds_byte =
        __builtin_amdgcn_readfirstlane((int)(uint32_t)(size_t)(lds_float*)lbuf);
    const uint64_t ga    = (uint64_t)(size_t)src;
    const uint32_t ga_lo = __builtin_amdgcn_readfirstlane((int)(uint32_t)(ga & 0xffffffffu));
    const uint32_t ga_hi = __builtin_amdgcn_readfirstlane((int)(uint32_t)(ga >> 32));
    const uint32_t rn    = __builtin_amdgcn_readfirstlane(n);

    // D# group 0 (ISA 8.3): count=1 | lds_addr | global_addr[56:0] | type=2
    u32x4 g0;
    g0[0] = 1u;
    g0[1] = lds_byte;
    g0[2] = ga_lo;
    g0[3] = (ga_hi & 0x1ffffffu) | (2u << 30);

    // D# group 1 (ISA 8.4): data_size=2 (4B); tensor_dim0=32; tensor_dim1=n;
    // tile_dim0=32; tile_dim1=n; tensor_dim0_stride=32 elements.
    i32x8 g1;
    g1[0] = (int)(2u << 16);                    // workgroup_mask=0, data_size=4B
    g1[1] = (int)(32u << 16);                   // tensor_dim0 lo16 @ [31:16]
    g1[2] = (int)((rn & 0xffffu) << 16);        // tensor_dim0 hi16=0 | tensor_dim1 lo16
    g1[3] = (int)(32u << 16);                   // tensor_dim1 hi16=0 | tile_dim0=32
    g1[4] = (int)(rn & 0xffffu);                // tile_dim1=n | tile_dim2=0
    g1[5] = 32;                                 // tensor_dim0_stride lo32
    g1[6] = 0;                                  // stride hi16 | dim1_stride lo16
    g1[7] = 0;

    i32x4 z4 = {0, 0, 0, 0};
#if __clang_major__ >= 23
    i32x8 z8 = {0, 0, 0, 0, 0, 0, 0, 0};
    __builtin_amdgcn_tensor_load_to_lds(g0, g1, z4, z4, z8, 0);
#else
    __builtin_amdgcn_tensor_load_to_lds(g0, g1, z4, z4, 0);
#endif
#else
    const int chunks = (n + 3) >> 2;
    const int sub    = lane >> 3;        // row within chunk 0..3
    const int qofs   = (lane & 7) << 2;  // 4-float group within row
    for (int c = 0; c < chunks; ++c) {
        const int row  = c * 4 + sub;
        const int srow = (row < n) ? row : (n - 1);  // clamp tail (dup data, never read)
        float4 v = *(const float4*)(src + (size_t)srow * NUM_CLASSES + qofs);
        *(float4*)(lbuf + row * NUM_CLASSES + qofs) = v;
    }
#endif
}

__device__ __forceinline__ void wait_tensor_le0() {
#if USE_TDM
    __builtin_amdgcn_s_wait_tensorcnt(0);
    __asm__ volatile("" ::: "memory");
#endif
}

__device__ __forceinline__ void wait_tensor_le1() {
#if USE_TDM
    __builtin_amdgcn_s_wait_tensorcnt(1);   // oldest outstanding TDM op retired
    __asm__ volatile("" ::: "memory");
#endif
}

// ---------------------------------------------------------------------------
// Main kernel (placed first for disasm inspection): one wave32 per image,
// lane = class. Persistent waves, depth-2 TDM pipeline over double-buffered
// LDS tiles: issue DMA for image k+1, wait tensorcnt<=1, compute image k.
// ---------------------------------------------------------------------------
__global__ __launch_bounds__(32 * WPB)
void CellLogitLSE_main_kernel(const float* __restrict__ logits,
                              const int*  __restrict__ counts,
                              const int*  __restrict__ offsets,
                              float*      __restrict__ out, int B) {
    __shared__ float tile[WPB][NBUF][MAX_ROWS * NUM_CLASSES];   // 64 KB / block

    const int wave   = threadIdx.x >> 5;
    const int lane   = threadIdx.x & 31;
    const int wid    = blockIdx.x * WPB + wave;
    const int stride = gridDim.x * WPB;

    if (wid >= B) return;

    // Preload first image's tile into buffer 0.
    {
        const int n0 = counts[wid];
        if (n0 > 0)
            stage_tile(logits + (size_t)offsets[wid] * NUM_CLASSES, n0,
                       &tile[wave][0][0], lane);
    }

    int parity = 0;
    for (int img = wid; img < B; img += stride, parity ^= 1) {
        const int n   = counts[img];
        const int nxt = img + stride;

        // Issue next image's DMA into the other buffer (freed last iteration).
        bool issued_next = false;
        if (nxt < B) {
            const int nn = counts[nxt];
            if (nn > 0) {
                stage_tile(logits + (size_t)offsets[nxt] * NUM_CLASSES, nn,
                           &tile[wave][parity ^ 1][0], lane);
                issued_next = true;
            }
        }

        float* o = out + (size_t)img * NUM_CLASSES;
        if (n <= 0) {                       // empty image -> zero row
            o[lane] = 0.0f;
            continue;
        }

        // TENSORcnt retires in order per wave: outstanding set here is
        // {load(img)} plus optionally {load(nxt)} just issued.
        if (issued_next) wait_tensor_le1();
        else             wait_tensor_le0();

        const float* lbuf = &tile[wave][parity][0];

        // Pass 1: per-class max over rows (lane -> distinct LDS bank).
        float m = -3.0e38f;
#pragma unroll 4
        for (int r = 0; r < n; ++r)
            m = fmaxf(m, lbuf[r * NUM_CLASSES + lane]);

        // Pass 2: s = sum_r exp2(SC * (x - m)),  SC = 5 * log2(e)
        const float SC  = 7.21347520444481703680f;
        const float msc = m * SC;
        float s = 0.0f;
#pragma unroll 4
        for (int r = 0; r < n; ++r) {
            float x = lbuf[r * NUM_CLASSES + lane];
            s += FAST_EXP2(__builtin_fmaf(x, SC, -msc));
        }

        // out = m + (ln2/5) * (log2(s) - log2(n))
        const float log2s = FAST_LOG2(s);
        const float log2n = FAST_LOG2((float)n);
        o[lane] = m + 0.13862943611198906188f * (log2s - log2n);
    }
}

// ---------------------------------------------------------------------------
// Kernel 1: exclusive prefix sum of cell_counts -> offsets (one workgroup)
// ---------------------------------------------------------------------------
__global__ __launch_bounds__(SCAN_THREADS)
void CellLogitLSE_scan_kernel(const int* __restrict__ counts,
                              int* __restrict__ offsets, int B) {
    __shared__ int lds[SCAN_THREADS];
    const int t   = threadIdx.x;
    const int per = (B + SCAN_THREADS - 1) / SCAN_THREADS;
    const int beg = t * per;
    const int end = (beg + per < B) ? (beg + per) : B;

    int sum = 0;
    for (int i = beg; i < end; ++i) sum += counts[i];
    lds[t] = sum;
    __syncthreads();

    // Hillis-Steele inclusive scan over the chunk sums
    for (int off = 1; off < SCAN_THREADS; off <<= 1) {
        int add = (t >= off) ? lds[t - off] : 0;
        __syncthreads();
        lds[t] += add;
        __syncthreads();
    }

    int run = lds[t] - sum;   // exclusive prefix of this chunk
    for (int i = beg; i < end; ++i) {
        offsets[i] = run;
        run += counts[i];
    }
}

// ---------------------------------------------------------------------------
extern "C" void kernel_launch(void* const* d_in, const int* in_sizes, int n_in,
                              void* d_out, int out_size, void* d_ws, size_t ws_size,
                              hipStream_t stream) {
    const float* logits = (const float*)d_in[0];
    const int*   counts = (const int*)d_in[1];
    const int    B      = in_sizes[1];

    int* offsets = (int*)d_ws;    // B ints of scratch

    CellLogitLSE_scan_kernel<<<1, SCAN_THREADS, 0, stream>>>(counts, offsets, B);

    CellLogitLSE_main_kernel<<<BLOCKS, 32 * WPB, 0, stream>>>(
        logits, counts, offsets, (float*)d_out, B);
}